// DSPFSeparableConv_35759897706916
// MI455X (gfx1250) — compile-verified
//
#include <hip/hip_runtime.h>
#include <math.h>

#define DILc 2
#define Bc 4
#define Cc 256
#define Oc 256
#define Gc 64
#define Hc 64
#define Wc 64
#define HWc (Hc * Wc)

typedef float v2f __attribute__((ext_vector_type(2)));
typedef float v8f __attribute__((ext_vector_type(8)));

// ---------------------------------------------------------------------------
// Kernel 1: offset branch: conv3x3(grad_feats[B,G,H,W], off_w[18,G,3,3],pad=1)
//           + BN(18) + ReLU  ->  off[B,18,H,W]
// ---------------------------------------------------------------------------
__global__ __launch_bounds__(256) void dspf_offset_conv(
    const float* __restrict__ gf, const float* __restrict__ ow,
    const float* __restrict__ gmm, const float* __restrict__ bta,
    const float* __restrict__ mu, const float* __restrict__ var,
    float* __restrict__ off) {
  const int idx = blockIdx.x * 256 + threadIdx.x;  // [B*18*H*W)
  const int w = idx & (Wc - 1);
  const int h = (idx >> 6) & (Hc - 1);
  const int j = (idx >> 12) % 18;
  const int b = idx / (18 * HWc);

  const float* gfb = gf + (size_t)b * Gc * HWc;
  const float* wj = ow + (size_t)j * Gc * 9;

  float acc = 0.f;
  for (int g = 0; g < Gc; ++g) {
    const float* gp = gfb + g * HWc;
    const float* wp = wj + g * 9;
#pragma unroll
    for (int ty = 0; ty < 3; ++ty) {
      const int hy = h + ty - 1;
      if (hy < 0 || hy >= Hc) continue;
#pragma unroll
      for (int tx = 0; tx < 3; ++tx) {
        const int wx = w + tx - 1;
        if (wx < 0 || wx >= Wc) continue;
        acc += gp[hy * Wc + wx] * wp[ty * 3 + tx];
      }
    }
  }
  const float sc = gmm[j] * rsqrtf(var[j] + 1e-5f);
  const float v = (acc - mu[j]) * sc + bta[j];
  off[idx] = v > 0.f ? v : 0.f;
}

// ---------------------------------------------------------------------------
// Kernel 2: x [B,C,H,W] -> xt [B,H,W,C]  (tiled LDS transpose, coalesced both
// directions) so bilinear gathers read 256 contiguous channels per corner.
// ---------------------------------------------------------------------------
__global__ __launch_bounds__(256) void dspf_transpose(
    const float* __restrict__ x, float* __restrict__ xt) {
  __shared__ float tile[32][33];
  const int b = blockIdx.z;
  const int p0 = blockIdx.x * 32;  // pixel index h*W+w
  const int c0 = blockIdx.y * 32;  // channel
  const int tx = threadIdx.x, ty = threadIdx.y;
#pragma unroll
  for (int i = ty; i < 32; i += 8)
    tile[i][tx] = x[((size_t)b * Cc + c0 + i) * HWc + p0 + tx];
  __syncthreads();
#pragma unroll
  for (int i = ty; i < 32; i += 8)
    xt[((size_t)b * HWc + p0 + i) * Cc + c0 + tx] = tile[tx][i];
}

// ---------------------------------------------------------------------------
// Kernel 3 (fused): per workgroup = 16 consecutive pixels (same b,h row).
// Phase 1: bilinear gather of 9 deformed taps over all 256 channels, reduced
//          by depthwise weight -> LDS inter[16][260] (padded, conflict-free).
// Phase 2: WMMA f32 16x16x4 GEMM vs pw_w (K=256 -> 64 chained v_wmma),
//          fused BN + ReLU, coalesced float4 stores.
// ---------------------------------------------------------------------------
__global__ __launch_bounds__(256) void dspf_fused(
    const float* __restrict__ xt,   // [B,H,W,C]
    const float* __restrict__ off,  // [B,18,H,W]
    const float* __restrict__ dw,   // [C,9]
    const float* __restrict__ pw,   // [O,C]
    const float* __restrict__ bg, const float* __restrict__ bb,
    const float* __restrict__ bm, const float* __restrict__ bv,
    float* __restrict__ out) {      // [B,O,H,W]
  __shared__ float inter[16][260];  // row stride 260: 16 lanes -> 16 banks

  const int t = threadIdx.x;
  const int m0 = blockIdx.x * 16;
  const int b = m0 / HWc;
  const int rem = m0 % HWc;
  const int h = rem / Wc;
  const int w0 = rem % Wc;  // multiple of 16 (W=64)

  // ---------------- Phase 1: gather + depthwise ----------------
  {
    const int p = t >> 4;           // pixel 0..15
    const int cq = (t & 15) * 16;   // 16-channel slab
    const int w = w0 + p;
    const float* offb = off + (size_t)b * 18 * HWc + h * Wc + w;

    float acc[16];
#pragma unroll
    for (int i = 0; i < 16; ++i) acc[i] = 0.f;

#pragma unroll
    for (int k = 0; k < 9; ++k) {
      const float dy = offb[(2 * k) * HWc];
      const float dx = offb[(2 * k + 1) * HWc];
      const float gy = (float)h + (float)((k / 3 - 1) * DILc) + dy;
      const float gx = (float)w + (float)((k % 3 - 1) * DILc) + dx;
      const float fy0 = floorf(gy), fx0 = floorf(gx);
      const int iy0 = (int)fy0, ix0 = (int)fx0;
      const int iy1 = iy0 + 1, ix1 = ix0 + 1;
      const float wy = gy - fy0, wx = gx - fx0;

      const bool vy0 = (iy0 >= 0) && (iy0 < Hc);
      const bool vy1 = (iy1 >= 0) && (iy1 < Hc);
      const bool vx0 = (ix0 >= 0) && (ix0 < Wc);
      const bool vx1 = (ix1 >= 0) && (ix1 < Wc);
      // fold validity masks into the bilinear weights
      const float m00 = (vy0 && vx0) ? (1.f - wy) * (1.f - wx) : 0.f;
      const float m01 = (vy0 && vx1) ? (1.f - wy) * wx : 0.f;
      const float m10 = (vy1 && vx0) ? wy * (1.f - wx) : 0.f;
      const float m11 = (vy1 && vx1) ? wy * wx : 0.f;

      const int cy0 = min(max(iy0, 0), Hc - 1), cy1 = min(max(iy1, 0), Hc - 1);
      const int cx0 = min(max(ix0, 0), Wc - 1), cx1 = min(max(ix1, 0), Wc - 1);
      const float* p00 = xt + (((size_t)b * Hc + cy0) * Wc + cx0) * Cc + cq;
      const float* p01 = xt + (((size_t)b * Hc + cy0) * Wc + cx1) * Cc + cq;
      const float* p10 = xt + (((size_t)b * Hc + cy1) * Wc + cx0) * Cc + cq;
      const float* p11 = xt + (((size_t)b * Hc + cy1) * Wc + cx1) * Cc + cq;

#pragma unroll
      for (int cc = 0; cc < 16; cc += 4) {
        const float4 a0 = *(const float4*)(p00 + cc);
        const float4 a1 = *(const float4*)(p01 + cc);
        const float4 a2 = *(const float4*)(p10 + cc);
        const float4 a3 = *(const float4*)(p11 + cc);
        const float s0 = m00 * a0.x + m01 * a1.x + m10 * a2.x + m11 * a3.x;
        const float s1 = m00 * a0.y + m01 * a1.y + m10 * a2.y + m11 * a3.y;
        const float s2 = m00 * a0.z + m01 * a1.z + m10 * a2.z + m11 * a3.z;
        const float s3 = m00 * a0.w + m01 * a1.w + m10 * a2.w + m11 * a3.w;
        acc[cc + 0] += s0 * dw[(cq + cc + 0) * 9 + k];
        acc[cc + 1] += s1 * dw[(cq + cc + 1) * 9 + k];
        acc[cc + 2] += s2 * dw[(cq + cc + 2) * 9 + k];
        acc[cc + 3] += s3 * dw[(cq + cc + 3) * 9 + k];
      }
    }
#pragma unroll
    for (int i = 0; i < 16; ++i) inter[p][cq + i] = acc[i];
  }
  __syncthreads();

  // ---------------- Phase 2: WMMA pointwise GEMM + BN + ReLU ----------------
  const int lane = t & 31;
  const int wid = t >> 5;        // wave 0..7
  const int half = lane >> 4;    // 0: K slots {k0,k0+1}, 1: {k0+2,k0+3}
  const int l16 = lane & 15;
  const float* interRow = &inter[l16][0];  // A row: M = l16

  for (int ti = 0; ti < 2; ++ti) {
    const int nt = wid + ti * 8;       // 16 N-tiles over 8 waves x2
    const int o = nt * 16 + l16;       // this lane's output channel (N)
    const float* pwRow = pw + (size_t)o * Cc;
    __builtin_prefetch(pwRow, 0, 1);   // global_prefetch_b8

    v8f c = {};
#pragma unroll 8
    for (int k0 = 0; k0 < Cc; k0 += 4) {
      const int kk = k0 + half * 2;
      const float2 av = *(const float2*)(interRow + kk);  // ds_load_b64
      const float2 bvv = *(const float2*)(pwRow + kk);    // global b64
      v2f a;  a.x = av.x;  a.y = av.y;
      v2f bmx; bmx.x = bvv.x; bmx.y = bvv.y;
      // D = A(16x4,f32) * B(4x16,f32) + C ; 64 chained per tile
      c = __builtin_amdgcn_wmma_f32_16x16x4_f32(
          false, a, false, bmx, (short)0, c, false, false);
    }

    const float sc = bg[o] * rsqrtf(bv[o] + 1e-5f);
    const float mu = bm[o];
    const float be = bb[o];
    float vals[8];
#pragma unroll
    for (int r = 0; r < 8; ++r) {  // VGPR r: M = r (lanes<16) / r+8 (lanes>=16)
      const float v = (c[r] - mu) * sc + be;
      vals[r] = v > 0.f ? v : 0.f;
    }
    float* dst = out + (((size_t)b * Oc + o) * Hc + h) * Wc + w0 + half * 8;
    *(float4*)(dst + 0) = make_float4(vals[0], vals[1], vals[2], vals[3]);
    *(float4*)(dst + 4) = make_float4(vals[4], vals[5], vals[6], vals[7]);
  }
}

// ---------------------------------------------------------------------------
extern "C" void kernel_launch(void* const* d_in, const int* in_sizes, int n_in,
                              void* d_out, int out_size, void* d_ws,
                              size_t ws_size, hipStream_t stream) {
  const float* x = (const float*)d_in[0];
  const float* gf = (const float*)d_in[1];
  const float* ow = (const float*)d_in[2];
  const float* og = (const float*)d_in[3];
  const float* ob = (const float*)d_in[4];
  const float* om = (const float*)d_in[5];
  const float* ov = (const float*)d_in[6];
  const float* dw = (const float*)d_in[7];
  const float* pw = (const float*)d_in[8];
  const float* bg = (const float*)d_in[9];
  const float* bbta = (const float*)d_in[10];
  const float* bm = (const float*)d_in[11];
  const float* bv = (const float*)d_in[12];
  float* out = (float*)d_out;

  float* off_ws = (float*)d_ws;                        // B*18*H*W floats
  float* xt_ws = off_ws + (size_t)Bc * 18 * HWc;       // B*H*W*C floats

  dspf_offset_conv<<<(Bc * 18 * HWc) / 256, 256, 0, stream>>>(
      gf, ow, og, ob, om, ov, off_ws);

  dim3 tb(32, 8, 1), tg(HWc / 32, Cc / 32, Bc);
  dspf_transpose<<<tg, tb, 0, stream>>>(x, xt_ws);

  dspf_fused<<<(Bc * HWc) / 16, 256, 0, stream>>>(
      xt_ws, off_ws, dw, pw, bg, bbta, bm, bv, out);
}